// CornerPool2d_58179626991849
// MI455X (gfx1250) — compile-verified
//
#include <hip/hip_runtime.h>
#include <stdint.h>

// ---------------------------------------------------------------------------
// CornerPool2d (TopLeft): out = reverse-cummax over H  +  reverse-cummax over W
// in:  [16, 256, 256, 64] f32 (NHWC, C contiguous)
// Memory-bound: 512MB min traffic -> ~22us @ 23.3 TB/s. Single fused pass,
// input read exactly once. CDNA5 async global->LDS prefetch (ASYNCcnt) keeps
// ~14MB in flight across the grid to hide HBM latency in the serial h-scan.
// ---------------------------------------------------------------------------

#if defined(__has_builtin)
#  if __has_builtin(__builtin_amdgcn_global_load_async_to_lds_b128) && \
      __has_builtin(__builtin_amdgcn_s_wait_asynccnt)
#    define USE_ASYNC 1
#  endif
#endif
#ifndef USE_ASYNC
#  define USE_ASYNC 0
#endif

typedef float f4 __attribute__((ext_vector_type(4)));

// Builtin signature (from hipcc diagnostic): pointee is vector_size(16) int.
typedef int i4v __attribute__((vector_size(16)));
typedef __attribute__((address_space(1))) i4v i4v_glob;  // global (AS1)
typedef __attribute__((address_space(3))) i4v i4v_lds;   // LDS (AS3, 32-bit)

#define NB 16
#define NH 256
#define NW 256
#define NC 64
#define CH 4                  // channels per workgroup (float4 per thread)
#define NCHUNK (NC / CH)      // 16 channel chunks
#define PF 15                 // rotating LDS row buffers (PF-1 rows in flight)

__device__ __forceinline__ f4 f4max(f4 a, f4 b) {
  f4 r;
  r.x = fmaxf(a.x, b.x);
  r.y = fmaxf(a.y, b.y);
  r.z = fmaxf(a.z, b.z);
  r.w = fmaxf(a.w, b.w);
  return r;
}

#if USE_ASYNC
__device__ __forceinline__ void issue_row_b128(const float* gp, const f4* lp) {
  // flat->AS1 is a numeric no-op; flat->AS3 is a truncation to the low 32 bits
  // (HW takes addr[31:0] as the LDS address).
  __builtin_amdgcn_global_load_async_to_lds_b128(
      (i4v_glob*)(uintptr_t)gp,
      (i4v_lds*)(uint32_t)(uintptr_t)lp,
      0, 0);
}
#endif

__global__ __launch_bounds__(NW) void corner_pool_fused(
    const float* __restrict__ in, float* __restrict__ out) {
  __shared__ f4 rowbuf[PF][NW];    // 15 * 4KB = 60KB rotating row buffers
  __shared__ f4 waveTot[2][8];     // per-wave row-suffix totals, dbl-buffered

  const int t     = threadIdx.x;          // w position
  const int b     = blockIdx.x >> 4;      // batch
  const int chunk = blockIdx.x & (NCHUNK - 1);
  const int lane  = t & 31;
  const int wid   = t >> 5;               // 8 waves per block

  const size_t strideH = (size_t)NW * NC;                       // 16384 floats
  const size_t base0   = (size_t)b * NH * strideH +
                         (size_t)t * NC + (size_t)(chunk * CH); // 16B aligned

#if USE_ASYNC
  // Prologue: put PF-1 rows in flight (rows 255 .. 255-(PF-2)).
#pragma unroll
  for (int i = 0; i < PF - 1; ++i) {
    issue_row_b128(in + base0 + (size_t)(NH - 1 - i) * strideH, &rowbuf[i][t]);
  }
#endif

  f4 colmax = {-__builtin_inff(), -__builtin_inff(),
               -__builtin_inff(), -__builtin_inff()};

  int buf = 0;  // buffer holding row h
  for (int h = NH - 1; h >= 0; --h) {
#if USE_ASYNC
    // Refill the buffer consumed last iteration with row h-(PF-1).
    const int hpre = h - (PF - 1);
    if (hpre >= 0) {
      const int bufpre = (buf == 0) ? (PF - 1) : (buf - 1);
      issue_row_b128(in + base0 + (size_t)hpre * strideH, &rowbuf[bufpre][t]);
    }
    // Async loads complete in order: waiting for <=PF-1 outstanding means the
    // oldest (row h) has landed. Tail rows have fewer outstanding -> wait 0.
    if (h >= PF - 1) {
      __builtin_amdgcn_s_wait_asynccnt(PF - 1);
    } else {
      __builtin_amdgcn_s_wait_asynccnt(0);
    }
    f4 v = rowbuf[buf][t];   // each thread reads the slot it loaded itself
#else
    f4 v = *(const f4*)(in + base0 + (size_t)h * strideH);
#endif
    buf = (buf + 1 == PF) ? 0 : buf + 1;

    // h-suffix max (register-resident, keyed by (w,c) owned by this thread)
    colmax = f4max(colmax, v);

    // w-suffix max: reverse inclusive max-scan across 256 threads.
    // Step 1: shuffle scan within wave32 (s[lane] = max over lanes lane..31).
    f4 s = v;
#pragma unroll
    for (int off = 1; off < 32; off <<= 1) {
      f4 o;
      o.x = __shfl_down(s.x, off, 32);
      o.y = __shfl_down(s.y, off, 32);
      o.z = __shfl_down(s.z, off, 32);
      o.w = __shfl_down(s.w, off, 32);
      if (lane + off < 32) s = f4max(s, o);
    }
    // Step 2: combine wave totals (lane 0 of each wave holds the wave max).
    const int par = h & 1;  // parity double-buffer => single barrier per row
    if (lane == 0) waveTot[par][wid] = s;
    __syncthreads();
    f4 suf = s;
#pragma unroll
    for (int j = 1; j < 8; ++j) {
      if (wid + j < 8) suf = f4max(suf, waveTot[par][wid + j]);
    }

    f4 res = colmax + suf;
    // Output is never re-read: non-temporal store keeps input resident in L2
    // for the sibling channel-chunk workgroups sharing the same cachelines.
    __builtin_nontemporal_store(res, (f4*)(out + base0 + (size_t)h * strideH));
  }
}

extern "C" void kernel_launch(void* const* d_in, const int* in_sizes, int n_in,
                              void* d_out, int out_size, void* d_ws,
                              size_t ws_size, hipStream_t stream) {
  (void)in_sizes; (void)n_in; (void)out_size; (void)d_ws; (void)ws_size;
  const float* in = (const float*)d_in[0];
  float* out = (float*)d_out;
  dim3 grid(NB * NCHUNK);  // 256 workgroups (adjacent ids share cachelines)
  dim3 block(NW);          // 256 threads = 8 wave32
  hipLaunchKernelGGL(corner_pool_fused, grid, block, 0, stream, in, out);
}